// FeatureTransformer_70136815944101
// MI455X (gfx1250) — compile-verified
//
#include <hip/hip_runtime.h>
#include <stdint.h>

typedef __attribute__((ext_vector_type(2))) float v2f;
typedef __attribute__((ext_vector_type(8))) float v8f;

#define NUM_INPUTS 40960
#define HIDDEN     1024
#define BATCH      8192
#define MAX_ACTIVE 32

#define WAVES     8
#define BT        16      // batch tile per block
#define HT        128     // hidden columns per wave
#define ROWSTRIDE 144     // padded LDS row stride (floats): bank-friendly, 16B aligned
#define STAGEF    (4 * ROWSTRIDE)          // floats per stage buffer (576)
#define STAGEB    (STAGEF * 4)             // bytes per stage buffer (2304)
#define NSTAGE    3                        // async pipeline depth
#define NCHUNK    128     // (16 batch * 32 feats) / 4 rows per chunk

__global__ __launch_bounds__(256)
void ft_embedbag_wmma(const int* __restrict__ act,
                      const float* __restrict__ W,
                      const float* __restrict__ bias,
                      float* __restrict__ out)
{
    __shared__ int   sidx[BT * MAX_ACTIVE];                  // 2 KB
    __shared__ float sbuf[WAVES][NSTAGE][STAGEF];            // 54 KB

    const int tid    = threadIdx.x;
    const int wid    = tid >> 5;
    const int lane   = tid & 31;
    const int lane15 = lane & 15;
    const int laneHi = lane >> 4;          // 0 or 1
    const int BB     = blockIdx.x * BT;
    const int HB     = wid * HT;

    // Stage the block's indices into LDS (chunk c uses sidx[4c..4c+3]).
    {
        const int* g = act + BB * MAX_ACTIVE;
        for (int i = tid; i < BT * MAX_ACTIVE; i += 256) sidx[i] = g[i];
    }
    __syncthreads();

    // Accumulators initialized with bias (C/D layout: b = v + 8*laneHi, h = HB + 16t + lane15).
    v8f acc[8];
#pragma unroll
    for (int t = 0; t < 8; ++t) {
        float bv = bias[HB + t * 16 + lane15];
#pragma unroll
        for (int v = 0; v < 8; ++v) acc[t][v] = bv;
    }

    const uint32_t gbase  = (uint32_t)HB * 4u + (uint32_t)lane * 16u;
    const uint32_t lwbase = (uint32_t)(uintptr_t)(&sbuf[wid][0][0]) + (uint32_t)lane * 16u;
    const int4*    sidx4  = (const int4*)sidx;

    // Async-gather one chunk (4 rows x 128 floats of this wave's h-slice) into stage `st`.
    auto prefetch = [&](int st, int4 rows) {
        uint32_t lbase = lwbase + (uint32_t)(st * STAGEB);
        uint32_t g0 = (uint32_t)rows.x * (uint32_t)(HIDDEN * 4) + gbase;
        uint32_t g1 = (uint32_t)rows.y * (uint32_t)(HIDDEN * 4) + gbase;
        uint32_t g2 = (uint32_t)rows.z * (uint32_t)(HIDDEN * 4) + gbase;
        uint32_t g3 = (uint32_t)rows.w * (uint32_t)(HIDDEN * 4) + gbase;
        asm volatile("global_load_async_to_lds_b128 %0, %1, %2"
                     :: "v"(lbase),                              "v"(g0), "s"(W) : "memory");
        asm volatile("global_load_async_to_lds_b128 %0, %1, %2"
                     :: "v"(lbase + (uint32_t)(1 * ROWSTRIDE * 4)), "v"(g1), "s"(W) : "memory");
        asm volatile("global_load_async_to_lds_b128 %0, %1, %2"
                     :: "v"(lbase + (uint32_t)(2 * ROWSTRIDE * 4)), "v"(g2), "s"(W) : "memory");
        asm volatile("global_load_async_to_lds_b128 %0, %1, %2"
                     :: "v"(lbase + (uint32_t)(3 * ROWSTRIDE * 4)), "v"(g3), "s"(W) : "memory");
    };

    // Prologue: fill stages 0 and 1; pipeline the int4 index load one chunk ahead.
    prefetch(0, sidx4[0]);
    prefetch(1, sidx4[1]);
    int4 rowsNext = sidx4[2];              // indices for chunk 2 (first in-loop prefetch)

    const int khalf = laneHi ? (2 * ROWSTRIDE) : 0;   // lanes 0-15 rows 0/1, lanes 16-31 rows 2/3
    int stC = 0;                            // stage of chunk c
    int stP = 2;                            // stage of chunk c+2

    for (int c = 0; c < NCHUNK; ++c) {
        if (c + 2 < NCHUNK) {
            prefetch(stP, rowsNext);
            int nc = c + 3 < NCHUNK ? c + 3 : NCHUNK - 1;
            rowsNext = sidx4[nc];           // latency hidden under this chunk's WMMAs
            asm volatile("s_wait_asynccnt 8" ::: "memory");   // chunk c's 4 loads done
        } else if (c + 1 < NCHUNK) {
            asm volatile("s_wait_asynccnt 4" ::: "memory");
        } else {
            asm volatile("s_wait_asynccnt 0" ::: "memory");
        }

        const int   bc  = c >> 3;                     // batch row this chunk feeds
        const float sel = (lane15 == bc) ? 1.0f : 0.0f;
        v2f a; a[0] = sel; a[1] = sel;                // one-hot A, constant over K

        const float* bufc = &sbuf[wid][stC][0];
#pragma unroll
        for (int t = 0; t < 8; ++t) {
            const int col = khalf + t * 16 + lane15;
            v2f b;
            b[0] = bufc[col];
            b[1] = bufc[col + ROWSTRIDE];
            // D[b,h] += (b==bc) * sum of the chunk's 4 gathered rows at column h
            acc[t] = __builtin_amdgcn_wmma_f32_16x16x4_f32(
                false, a, false, b, (short)0, acc[t], false, false);
        }

        stC = (stC == NSTAGE - 1) ? 0 : stC + 1;
        stP = (stP == NSTAGE - 1) ? 0 : stP + 1;
    }

    // Store: half-wave writes 64B contiguous segments.
    float* op = out + (size_t)(BB + laneHi * 8) * HIDDEN + HB + lane15;
#pragma unroll
    for (int v = 0; v < 8; ++v) {
#pragma unroll
        for (int t = 0; t < 8; ++t) {
            op[(size_t)v * HIDDEN + t * 16] = acc[t][v];
        }
    }
}

extern "C" void kernel_launch(void* const* d_in, const int* in_sizes, int n_in,
                              void* d_out, int out_size, void* d_ws, size_t ws_size,
                              hipStream_t stream) {
    (void)in_sizes; (void)n_in; (void)out_size; (void)d_ws; (void)ws_size;
    const int*   act  = (const int*)d_in[0];
    const float* W    = (const float*)d_in[1];
    const float* bias = (const float*)d_in[2];
    float*       out  = (float*)d_out;

    dim3 grid(BATCH / BT);   // 512 blocks
    dim3 block(256);         // 8 wave32 waves, one 16x128 tile each
    hipLaunchKernelGGL(ft_embedbag_wmma, grid, block, 0, stream, act, W, bias, out);
}